// HoGEdgeGateConv_72945724555746
// MI455X (gfx1250) — compile-verified
//
#include <hip/hip_runtime.h>
#include <hip/hip_bf16.h>
#include <math.h>

#define PI_F 3.14159265358979323846f

typedef float v2f __attribute__((ext_vector_type(2)));
typedef float v8f __attribute__((ext_vector_type(8)));

// ---------------------------------------------------------------------------
// 1) Channel mean over 64 channels: x[4,64,384,384] -> xmean[4,384,384]
// ---------------------------------------------------------------------------
__global__ __launch_bounds__(256) void mean_kernel(const float* __restrict__ X,
                                                   float* __restrict__ M) {
  int idx = blockIdx.x * 256 + threadIdx.x;  // exactly 4*147456 threads
  int b = idx / 147456;
  int p = idx % 147456;
  const float* x = X + (size_t)b * 64 * 147456 + p;
  float s = 0.f;
#pragma unroll
  for (int c = 0; c < 64; ++c) s += x[(size_t)c * 147456];
  M[idx] = s * (1.0f / 64.0f);
}

// ---------------------------------------------------------------------------
// 2) HoG feature per "cell". Faithful to reference row-major reshape:
//    cell (i,j) of patch bp = 64 pixels at row 8i + j/3, cols (j%3)*64..+63.
//    Sobel with per-patch zero padding; integer LDS histogram (deterministic).
//    hog[bp][bin][i][j], 16*9*24*24
// ---------------------------------------------------------------------------
__global__ __launch_bounds__(64) void hog_kernel(const float* __restrict__ XM,
                                                 float* __restrict__ HOG) {
  __shared__ int hist[9];
  const int blk = blockIdx.x;  // 16*576
  const int bp = blk / 576;
  const int rem = blk % 576;
  const int i = rem / 24, j = rem % 24;
  const int t = threadIdx.x;  // 64 threads = 64 pixels of this cell
  if (t < 9) hist[t] = 0;
  __syncthreads();
  const int ph = 8 * i + j / 3;
  const int pw = (j % 3) * 64 + t;
  const int b = bp & 3, wg = (bp >> 2) & 1, hg = bp >> 3;
  const float* base = XM + (size_t)b * 147456;
  float win[3][3];
#pragma unroll
  for (int di = 0; di < 3; ++di)
#pragma unroll
    for (int dj = 0; dj < 3; ++dj) {
      int r = ph + di - 1, cc = pw + dj - 1;
      win[di][dj] = (r >= 0 && r < 192 && cc >= 0 && cc < 192)
                        ? base[(size_t)(hg * 192 + r) * 384 + (wg * 192 + cc)]
                        : 0.f;
    }
  float dx = (win[0][2] - win[0][0]) + 2.f * (win[1][2] - win[1][0]) +
             (win[2][2] - win[2][0]);
  float dy = (win[2][0] - win[0][0]) + 2.f * (win[2][1] - win[0][1]) +
             (win[2][2] - win[0][2]);
  float dir = fabsf(atan2f(dy, dx));
  int bin = (int)floorf(dir / (PI_F / 9.f));
  bin = bin < 0 ? 0 : (bin > 8 ? 8 : bin);
  atomicAdd(&hist[bin], 1);
  __syncthreads();
  if (t < 9)
    HOG[(size_t)bp * 5184 + t * 576 + rem] =
        ((float)hist[t] * (1.f / 64.f)) * ((2 * t + 1) * (PI_F / 18.f));
}

// ---------------------------------------------------------------------------
// 3) hog_feat pipeline per patch: 1x1 (9->64) + dw3x3 (recomputed twice),
//    GroupNorm(8 groups), ReLU, global mean -> s[bp][64]. One block per bp.
//    Thread t handles channel c=t&63, quarter q=t>>6 (144 cells each).
// ---------------------------------------------------------------------------
__global__ __launch_bounds__(256) void hogfeat_kernel(
    const float* __restrict__ HOG, const float* __restrict__ W1,
    const float* __restrict__ B1, const float* __restrict__ DW,
    const float* __restrict__ GG, const float* __restrict__ GB,
    float* __restrict__ Sout) {
  __shared__ float hog_s[9 * 576];
  __shared__ float part[256];
  __shared__ float part2[256];
  __shared__ float gm[8], gr[8];
  const int bp = blockIdx.x;
  const int t = threadIdx.x;
  for (int i = t; i < 9 * 576; i += 256) hog_s[i] = HOG[(size_t)bp * 5184 + i];
  const int c = t & 63;
  const int q = t >> 6;
  float w1r[9], dwr[9];
#pragma unroll
  for (int k = 0; k < 9; ++k) {
    w1r[k] = W1[c * 9 + k];
    dwr[k] = DW[c * 9 + k];
  }
  const float b1c = B1[c];
  __syncthreads();

  auto conv1 = [&](int i, int j) -> float {
    if (i < 0 || i >= 24 || j < 0 || j >= 24) return 0.f;  // pad after bias
    float s = b1c;
#pragma unroll
    for (int k = 0; k < 9; ++k) s += w1r[k] * hog_s[k * 576 + i * 24 + j];
    return s;
  };
  auto dwval = [&](int i, int j) -> float {
    float s = 0.f;
#pragma unroll
    for (int di = 0; di < 3; ++di)
#pragma unroll
      for (int dj = 0; dj < 3; ++dj)
        s += dwr[di * 3 + dj] * conv1(i + di - 1, j + dj - 1);
    return s;
  };

  float ls = 0.f, lq = 0.f;
  for (int cell = q * 144; cell < q * 144 + 144; ++cell) {
    float v = dwval(cell / 24, cell % 24);
    ls += v;
    lq += v * v;
  }
  part[t] = ls;
  part2[t] = lq;
  __syncthreads();
  if (t < 8) {  // group t: channels 8t..8t+7, fixed-order sum over 32 partials
    float s = 0.f, sq = 0.f;
    for (int cc = t * 8; cc < t * 8 + 8; ++cc)
      for (int qq = 0; qq < 4; ++qq) {
        s += part[qq * 64 + cc];
        sq += part2[qq * 64 + cc];
      }
    float mean = s / 4608.f;
    float var = sq / 4608.f - mean * mean;
    gm[t] = mean;
    gr[t] = rsqrtf(var + 1e-5f);
  }
  __syncthreads();
  const int g = c >> 3;
  const float m = gm[g], r = gr[g];
  const float gamma = GG[c], beta = GB[c];
  float cs = 0.f;
  for (int cell = q * 144; cell < q * 144 + 144; ++cell) {
    float v = (dwval(cell / 24, cell % 24) - m) * r * gamma + beta;
    cs += fmaxf(v, 0.f);
  }
  part[t] = cs;
  __syncthreads();
  if (t < 64) {
    float s = 0.f;
    for (int qq = 0; qq < 4; ++qq) s += part[qq * 64 + t];
    Sout[bp * 64 + t] = s / 576.f;
  }
}

// ---------------------------------------------------------------------------
// 4) 1x1 conv as WMMA f32 GEMM. One wave = one 16-pixel tile x all Cout.
//    A (weights, 16x4 f32 layout): lane ln holds M=ln; VGPR0=K=k0+2*hi,
//    VGPR1=K=k0+2*hi+1. B (activations, 4x16): lane = pixel, same K split.
//    D (16x16 f32): VGPR v, lane -> (M=v+8*hi, N=ln). 16 K-steps, MT m-tiles.
//    PATCHED: read x through image2patches mapping. ADDS: add s[bp][k] to
//    input (t = xp + xh). STATS: emit per-block GroupNorm partial sums
//    (fixed-order shuffle tree -> LDS -> unique global slot; deterministic).
// ---------------------------------------------------------------------------
template <int CIN, int COUT, bool PATCHED, bool ADDS, bool STATS>
__global__ __launch_bounds__(256) void gemm1x1(
    const float* __restrict__ X, const float* __restrict__ W,
    const float* __restrict__ Bi, const float* __restrict__ S,
    float* __restrict__ Y, float* __restrict__ part, int HW,
    int blocksPerBatch) {
  constexpr int MT = COUT / 16;
  __shared__ float wlds[COUT * (CIN + 1)];  // stride CIN+1: no bank conflicts
  __shared__ float blds[COUT];
  __shared__ float slds[ADDS ? CIN : 1];
  __shared__ float red[8][16];  // wave x (group*2)

  const int tid = threadIdx.x;
  const int bp = blockIdx.y;

  for (int i = tid; i < COUT * CIN; i += 256)
    wlds[(i / CIN) * (CIN + 1) + (i % CIN)] = W[i];
  if (tid < COUT) blds[tid] = Bi[tid];
  if (ADDS) {
    if (tid < CIN) slds[tid] = S[bp * CIN + tid];
  }
  __syncthreads();

  const int wave = tid >> 5;
  const int lane = tid & 31;
  const int hi = lane >> 4;
  const int ln = lane & 15;

  const int tile = blockIdx.x * 8 + wave;
  const int pbase = tile * 16;
  const int p = pbase + ln;

  const float* src;
  size_t cstride;
  if (PATCHED) {
    const int b = bp & 3;
    const int wg = (bp >> 2) & 1;
    const int hg = bp >> 3;
    const int ph = p / 192, pw = p % 192;
    src = X + (size_t)b * CIN * 147456 + (size_t)(hg * 192 + ph) * 384 +
          (wg * 192 + pw);
    cstride = 147456;
  } else {
    src = X + ((size_t)bp * CIN) * HW + p;
    cstride = (size_t)HW;
  }

  v8f acc[MT] = {};
#pragma unroll
  for (int k0 = 0; k0 < CIN; k0 += 4) {
    const int kk = k0 + 2 * hi;
    v2f bm;
    bm.x = src[(size_t)kk * cstride];
    bm.y = src[(size_t)(kk + 1) * cstride];
    if (ADDS) {
      bm.x += slds[kk];
      bm.y += slds[kk + 1];
    }
#pragma unroll
    for (int mt = 0; mt < MT; ++mt) {
      v2f am;
      am.x = wlds[(mt * 16 + ln) * (CIN + 1) + kk];
      am.y = wlds[(mt * 16 + ln) * (CIN + 1) + kk + 1];
      acc[mt] = __builtin_amdgcn_wmma_f32_16x16x4_f32(
          false, am, false, bm, (short)0, acc[mt], false, false);
    }
  }

#pragma unroll
  for (int mt = 0; mt < MT; ++mt) {
    float lsum = 0.f, lsq = 0.f;
#pragma unroll
    for (int v = 0; v < 8; ++v) {
      const int m = mt * 16 + v + 8 * hi;
      float val = acc[mt][v] + blds[m];
      Y[((size_t)bp * COUT + m) * HW + pbase + ln] = val;
      if (STATS) {
        lsum += val;
        lsq += val * val;
      }
    }
    if (STATS) {
      // reduce across the 16 lanes of each half (masks 1,2,4,8 stay in-half)
#pragma unroll
      for (int off = 1; off < 16; off <<= 1) {
        lsum += __shfl_xor(lsum, off, 32);
        lsq += __shfl_xor(lsq, off, 32);
      }
      if (ln == 0) {
        const int g = 2 * mt + hi;  // channel group of 8
        red[wave][g * 2 + 0] = lsum;
        red[wave][g * 2 + 1] = lsq;
      }
    }
  }
  if (STATS) {
    __syncthreads();
    if (tid < 16) {
      float s = 0.f;
#pragma unroll
      for (int w = 0; w < 8; ++w) s += red[w][tid];
      part[((size_t)bp * blocksPerBatch + blockIdx.x) * 16 + tid] = s;
    }
  }
}

// ---------------------------------------------------------------------------
// 5) Depthwise: c<32 -> (1,3) on u[c]; c>=32 -> (3,1) on u[c-32]; zero pad.
//    u:[NB,32,H,W] -> v:[NB,64,H,W]
// ---------------------------------------------------------------------------
__global__ __launch_bounds__(256) void dw_kernel(
    const float* __restrict__ U, const float* __restrict__ WW,
    const float* __restrict__ WBv, const float* __restrict__ WH,
    const float* __restrict__ HBv, float* __restrict__ V, int NB, int H,
    int W) {
  size_t idx = (size_t)blockIdx.x * 256 + threadIdx.x;
  const size_t hw = (size_t)H * W;
  int p = (int)(idx % hw);
  int c = (int)((idx / hw) & 63);
  int nb = (int)(idx / (64 * hw));
  int ph = p / W, pw = p % W;
  float out;
  if (c < 32) {
    const float* u = U + ((size_t)nb * 32 + c) * hw + (size_t)ph * W;
    float a = (pw > 0) ? u[pw - 1] : 0.f;
    float b = u[pw];
    float d = (pw < W - 1) ? u[pw + 1] : 0.f;
    out = WBv[c] + a * WW[c * 3] + b * WW[c * 3 + 1] + d * WW[c * 3 + 2];
  } else {
    int ch = c - 32;
    const float* u = U + ((size_t)nb * 32 + ch) * hw;
    float a = (ph > 0) ? u[(size_t)(ph - 1) * W + pw] : 0.f;
    float b = u[(size_t)ph * W + pw];
    float d = (ph < H - 1) ? u[(size_t)(ph + 1) * W + pw] : 0.f;
    out = HBv[ch] + a * WH[ch * 3] + b * WH[ch * 3 + 1] + d * WH[ch * 3 + 2];
  }
  V[idx] = out;
}

// ---------------------------------------------------------------------------
// 6) Finalize GN stats from per-block partials (fixed order, deterministic).
//    st[(bp*8+g)*2] = mean, +1 = rstd
// ---------------------------------------------------------------------------
__global__ void reduce_stats(const float* __restrict__ part,
                             float* __restrict__ st, int blocksPerBatch,
                             float invN) {
  if (threadIdx.x != 0) return;
  int bp = blockIdx.x >> 3;
  int g = blockIdx.x & 7;
  float s = 0.f, q = 0.f;
  for (int blk = 0; blk < blocksPerBatch; ++blk) {
    const float* pp = part + ((size_t)bp * blocksPerBatch + blk) * 16 + g * 2;
    s += pp[0];
    q += pp[1];
  }
  float mean = s * invN;
  float var = q * invN - mean * mean;
  st[(bp * 8 + g) * 2] = mean;
  st[(bp * 8 + g) * 2 + 1] = rsqrtf(var + 1e-5f);
}

// ---------------------------------------------------------------------------
// 7) Gate: y1 = patches2image(sigmoid(GN_w(e)) * GN_c(f)) + x  (image layout)
// ---------------------------------------------------------------------------
__global__ __launch_bounds__(256) void gate_kernel(
    const float* __restrict__ E, const float* __restrict__ F,
    const float* __restrict__ X, const float* __restrict__ stW,
    const float* __restrict__ stC, const float* __restrict__ WG,
    const float* __restrict__ WB2, const float* __restrict__ CG,
    const float* __restrict__ CB2, float* __restrict__ Y) {
  size_t idx = (size_t)blockIdx.x * 256 + threadIdx.x;  // 4*64*147456
  int p = (int)(idx % 147456);
  int c = (int)((idx / 147456) & 63);
  int b = (int)(idx / (147456ull * 64));
  int h = p / 384, w = p % 384;
  int hg = h / 192, ph = h % 192, wg = w / 192, pw = w % 192;
  int bp = hg * 8 + wg * 4 + b;
  size_t pi = ((size_t)bp * 64 + c) * 36864 + ph * 192 + pw;
  int g = c >> 3;
  float mw = stW[(bp * 8 + g) * 2], rw = stW[(bp * 8 + g) * 2 + 1];
  float mc = stC[(bp * 8 + g) * 2], rc = stC[(bp * 8 + g) * 2 + 1];
  float x1 = (E[pi] - mw) * rw * WG[c] + WB2[c];
  x1 = 1.f / (1.f + __expf(-x1));
  float x2 = (F[pi] - mc) * rc * CG[c] + CB2[c];
  Y[idx] = x1 * x2 + X[idx];
}

// ---------------------------------------------------------------------------
// 8) Final GN apply: out = GN_f(e2)
// ---------------------------------------------------------------------------
__global__ __launch_bounds__(256) void final_kernel(
    const float* __restrict__ E2, const float* __restrict__ stF,
    const float* __restrict__ FG, const float* __restrict__ FB,
    float* __restrict__ OUT) {
  size_t idx = (size_t)blockIdx.x * 256 + threadIdx.x;  // 4*64*147456
  int c = (int)((idx / 147456) & 63);
  int b = (int)(idx / (147456ull * 64));
  int g = c >> 3;
  float m = stF[(b * 8 + g) * 2], r = stF[(b * 8 + g) * 2 + 1];
  OUT[idx] = (E2[idx] - m) * r * FG[c] + FB[c];
}

// ---------------------------------------------------------------------------
extern "C" void kernel_launch(void* const* d_in, const int* in_sizes, int n_in,
                              void* d_out, int out_size, void* d_ws,
                              size_t ws_size, hipStream_t stream) {
  (void)in_sizes;
  (void)n_in;
  (void)out_size;
  (void)ws_size;
  // inputs: x, then params sorted by name
  const float* x = (const float*)d_in[0];
  const float* c_b = (const float*)d_in[1];
  const float* c_gn_b = (const float*)d_in[2];
  const float* c_gn_g = (const float*)d_in[3];
  const float* c_w = (const float*)d_in[4];
  const float* f_gn_b = (const float*)d_in[5];
  const float* f_gn_g = (const float*)d_in[6];
  const float* f_h_b = (const float*)d_in[7];
  const float* f_h_w = (const float*)d_in[8];
  const float* f_in_b = (const float*)d_in[9];
  const float* f_in_w = (const float*)d_in[10];
  const float* f_out_b = (const float*)d_in[11];
  const float* f_out_w = (const float*)d_in[12];
  const float* f_w_b = (const float*)d_in[13];
  const float* f_w_w = (const float*)d_in[14];
  const float* hf1_b = (const float*)d_in[15];
  const float* hf1_w = (const float*)d_in[16];
  const float* hf_dw_w = (const float*)d_in[17];
  const float* hf_gn_b = (const float*)d_in[18];
  const float* hf_gn_g = (const float*)d_in[19];
  const float* w_gn_b = (const float*)d_in[20];
  const float* w_gn_g = (const float*)d_in[21];
  const float* w_h_b = (const float*)d_in[22];
  const float* w_h_w = (const float*)d_in[23];
  const float* w_in_b = (const float*)d_in[24];
  const float* w_in_w = (const float*)d_in[25];
  const float* w_out_b = (const float*)d_in[26];
  const float* w_out_w = (const float*)d_in[27];
  const float* w_w_b = (const float*)d_in[28];
  const float* w_w_w = (const float*)d_in[29];
  float* out = (float*)d_out;

  // workspace arena (floats), 64-element aligned
  float* ws = (float*)d_ws;
  size_t off = 0;
  auto take = [&](size_t n) {
    float* p = ws + off;
    off += (n + 63) & ~(size_t)63;
    return p;
  };
  float* xmean = take(589824);     // [4,384,384]
  float* hog = take(82944);        // [16,9,24,24]
  float* svec = take(1024);        // [16,64]
  float* partW = take(73728);      // [16,288,16]
  float* partC = take(73728);
  float* partF = take(73728);      // [4,1152,16]
  float* stW = take(256);
  float* stC = take(256);
  float* stF = take(64);
  float* buf1 = take(37748736);    // f  -> v2
  float* buf2 = take(37748736);    // v  -> y1
  float* buf3 = take(37748736);    // e  -> e2
  float* buf4 = take(18874368);    // u  -> u2

  // Stage A: HoG scale vector
  mean_kernel<<<2304, 256, 0, stream>>>(x, xmean);
  hog_kernel<<<9216, 64, 0, stream>>>(xmean, hog);
  hogfeat_kernel<<<16, 256, 0, stream>>>(hog, hf1_w, hf1_b, hf_dw_w, hf_gn_g,
                                         hf_gn_b, svec);

  // Stage B: edge_conv 'w' on (xp + s), c-conv on xp, GN stats
  gemm1x1<64, 32, true, true, false><<<dim3(288, 16), 256, 0, stream>>>(
      x, w_in_w, w_in_b, svec, buf4, nullptr, 36864, 288);
  dw_kernel<<<147456, 256, 0, stream>>>(buf4, w_w_w, w_w_b, w_h_w, w_h_b, buf2,
                                        16, 192, 192);
  gemm1x1<64, 64, false, false, true><<<dim3(288, 16), 256, 0, stream>>>(
      buf2, w_out_w, w_out_b, nullptr, buf3, partW, 36864, 288);
  gemm1x1<64, 64, true, false, true><<<dim3(288, 16), 256, 0, stream>>>(
      x, c_w, c_b, nullptr, buf1, partC, 36864, 288);
  reduce_stats<<<128, 32, 0, stream>>>(partW, stW, 288, 1.f / (8 * 36864));
  reduce_stats<<<128, 32, 0, stream>>>(partC, stC, 288, 1.f / (8 * 36864));
  gate_kernel<<<147456, 256, 0, stream>>>(buf3, buf1, x, stW, stC, w_gn_g,
                                          w_gn_b, c_gn_g, c_gn_b, buf2);

  // Stage C: edge_conv 'f' on y1, final GroupNorm
  gemm1x1<64, 32, false, false, false><<<dim3(1152, 4), 256, 0, stream>>>(
      buf2, f_in_w, f_in_b, nullptr, buf4, nullptr, 147456, 1152);
  dw_kernel<<<147456, 256, 0, stream>>>(buf4, f_w_w, f_w_b, f_h_w, f_h_b, buf1,
                                        4, 384, 384);
  gemm1x1<64, 64, false, false, true><<<dim3(1152, 4), 256, 0, stream>>>(
      buf1, f_out_w, f_out_b, nullptr, buf3, partF, 147456, 1152);
  reduce_stats<<<32, 32, 0, stream>>>(partF, stF, 1152, 1.f / (8 * 147456));
  final_kernel<<<147456, 256, 0, stream>>>(buf3, stF, f_gn_g, f_gn_b, out);
}